// RadonBackward_2817498546707
// MI455X (gfx1250) — compile-verified
//
#include <hip/hip_runtime.h>
#include <hip/hip_bf16.h>
#include <cstdint>

// Problem constants (from reference setup_inputs): B=4, C=1, A=180, P=384, N=256
#define A_ANG 180
#define PDET  384
#define NSZ   256
#define BATCH 4

#define CH          20                    // angles per LDS chunk
#define NCH         (A_ANG / CH)          // 9 chunks
#define CHUNK_FLOATS (CH * PDET)          // 7680 floats = 30720 B
// b64 units per chunk = 30720/8 = 3840 = 15 * 256 threads  -> uniform 15 loads/thread

static __device__ __forceinline__ unsigned lds_addr_of(const void* p) {
    // generic pointer to __shared__: low 32 bits are the LDS byte offset
    return (unsigned)(size_t)p;
}

static __device__ __forceinline__ void wait_async_0() {
#if __has_builtin(__builtin_amdgcn_s_wait_asynccnt)
    __builtin_amdgcn_s_wait_asynccnt(0);
#else
    asm volatile("s_wait_asynccnt 0" ::: "memory");
#endif
}

static __device__ __forceinline__ void wait_async_15() {
#if __has_builtin(__builtin_amdgcn_s_wait_asynccnt)
    __builtin_amdgcn_s_wait_asynccnt(15);
#else
    asm volatile("s_wait_asynccnt 15" ::: "memory");
#endif
}

// Issue async copy of one chunk (CH rows of the sinogram) into LDS.
// 15 x b64 per thread, uniform across the wave so ASYNCcnt bookkeeping is exact.
static __device__ __forceinline__ void issue_chunk_async(
    const float* __restrict__ gsrc, float* lbase, int tid)
{
#pragma unroll
    for (int u = 0; u < 15; ++u) {
        const int fidx = (u * 256 + tid) * 2;   // float index, 2 floats per b64
        unsigned laddr = lds_addr_of(lbase + fidx);
        unsigned long long ga = (unsigned long long)(size_t)(gsrc + fidx);
        asm volatile("global_load_async_to_lds_b64 %0, %1, off"
                     :: "v"(laddr), "v"(ga)
                     : "memory");
    }
}

__global__ __launch_bounds__(256)
void radon_bp_kernel(const float* __restrict__ sino,
                     const float* __restrict__ thetas,
                     const float* __restrict__ positions,
                     float* __restrict__ out)
{
    __shared__ __align__(16) float sino_s[2 * CHUNK_FLOATS]; // 60 KB double buffer
    __shared__ float cosT[A_ANG];
    __shared__ float sinT[A_ANG];

    const int tid = threadIdx.x;
    const int b   = blockIdx.z;

    // Build sincos table cooperatively (disjoint LDS region from sino_s).
    for (int a = tid; a < A_ANG; a += 256) {
        float s, c;
        sincosf(thetas[a], &s, &c);
        cosT[a] = c;
        sinT[a] = s;
    }

    const float p0    = positions[0];
    const float invdp = 1.0f / (positions[1] - positions[0]);
    const float q     = -p0 * invdp;

    // Thread -> pixel mapping: 64(x) x 16(y) tile, 2x2 strided pixels per thread.
    const int tx  = tid & 31;
    const int ty  = tid >> 5;
    const int ix0 = blockIdx.x * 64 + tx;      // second x pixel: ix0 + 32
    const int iy0 = blockIdx.y * 16 + ty;      // second y pixel: iy0 + 8
    const float xs0 = (float)ix0 - 127.5f;     // (N-1)/2 = 127.5
    const float ys0 = 127.5f - (float)iy0;

    const float* src = sino + (size_t)b * (A_ANG * PDET);

    // Prologue: kick off chunk 0 into buffer 0.
    issue_chunk_async(src, &sino_s[0], tid);

    float acc00 = 0.0f, acc10 = 0.0f, acc01 = 0.0f, acc11 = 0.0f;

    for (int k = 0; k < NCH; ++k) {
        const int cb = (k & 1) * CHUNK_FLOATS;

        if (k + 1 < NCH) {
            // Issue next chunk into the other buffer, then drain the previous
            // chunk's 15 loads (leave the 15 just-issued in flight).
            issue_chunk_async(src + (k + 1) * CHUNK_FLOATS,
                              &sino_s[((k + 1) & 1) * CHUNK_FLOATS], tid);
            wait_async_15();
        } else {
            wait_async_0();
        }
        __syncthreads();   // all waves have chunk k resident (also fences sincos table)

        const int a0 = k * CH;
        for (int aa = 0; aa < CH; ++aa) {
            const float cth = cosT[a0 + aa];
            const float sth = sinT[a0 + aa];
            const float t0  = fmaf(xs0, cth, ys0 * sth);
            const float fb  = fmaf(t0, invdp, q);           // f at (ix0, iy0)
            const float fX  = (32.0f * invdp) * cth;        // dx = +32
            const float fY  = (-8.0f * invdp) * sth;        // dy = +8 -> ys -= 8
            const int   rb  = cb + aa * PDET;

            auto ACC = [&](float& acc, float f) {
                const float fl = floorf(f);
                const int   i0 = (int)fl;
                const float w  = f - fl;
                const int   ic = min(max(i0, 0), PDET - 2);
                const float v0 = sino_s[rb + ic];
                const float v1 = sino_s[rb + ic + 1];
                const float val = fmaf(w, v1 - v0, v0);
                acc += (i0 >= 0 && i0 <= PDET - 2) ? val : 0.0f;
            };

            ACC(acc00, fb);
            ACC(acc10, fb + fX);
            ACC(acc01, fb + fY);
            ACC(acc11, fb + fX + fY);
        }
        __syncthreads();   // everyone done reading buf[k&1] before it is overwritten
    }

    float* o = out + (size_t)b * (NSZ * NSZ);
    o[iy0 * NSZ + ix0]            = acc00;
    o[iy0 * NSZ + ix0 + 32]       = acc10;
    o[(iy0 + 8) * NSZ + ix0]      = acc01;
    o[(iy0 + 8) * NSZ + ix0 + 32] = acc11;
}

extern "C" void kernel_launch(void* const* d_in, const int* in_sizes, int n_in,
                              void* d_out, int out_size, void* d_ws, size_t ws_size,
                              hipStream_t stream) {
    (void)in_sizes; (void)n_in; (void)d_ws; (void)ws_size; (void)out_size;
    const float* sino      = (const float*)d_in[0];   // (4,1,180,384) f32
    const float* thetas    = (const float*)d_in[1];   // (180,) f32
    const float* positions = (const float*)d_in[2];   // (384,) f32
    float* out             = (float*)d_out;           // (4,1,256,256) f32

    dim3 grid(NSZ / 64, NSZ / 16, BATCH);  // (4, 16, 4)
    dim3 block(256);
    radon_bp_kernel<<<grid, block, 0, stream>>>(sino, thetas, positions, out);
}